// HeadlessEdgeRegressionModel_39247411151001
// MI455X (gfx1250) — compile-verified
//
#include <hip/hip_runtime.h>
#include <cstddef>

typedef __attribute__((ext_vector_type(2))) float v2f;
typedef __attribute__((ext_vector_type(8))) float v8f;

#define DIM 128

// ---------------------------------------------------------------------------
// Core WMMA tile loop: C[NT tiles of 16x16] += A_lds[NT*16 x K] * W[K x 128]
// A_lds has row stride K+4 floats (bank-conflict-free fragment reads).
// Each wave handles column strip n0..n0+15; B fragment shared across NT tiles.
// f32 WMMA layout (ISA 7.12.2):
//   A 16x4 : lanes 0-15 -> M=lane, {K=kk,kk+1}; lanes 16-31 -> {K=kk+2,kk+3}
//   B 4x16 : lanes 0-15 -> N=lane, {K=kk,kk+1}; lanes 16-31 -> {K=kk+2,kk+3}
//   C/D    : VGPR i -> (M=i, N=lane) lanes 0-15 ; (M=i+8, N=lane-16) lanes 16-31
// ---------------------------------------------------------------------------
template<int K, int NT>
__device__ __forceinline__ void mma_tiles(const float* __restrict__ As,
                                          const float* __restrict__ W,
                                          v8f acc[NT], int n0, int lane) {
  const int half = lane >> 4;       // 0: K pair {0,1}, 1: K pair {2,3}
  const int l16  = lane & 15;
  const int koff = half * 2;
  constexpr int LDK = K + 4;
  for (int kk = 0; kk < K; kk += 4) {
    const int bk = kk + koff;
    v2f b;
    b.x = W[bk * DIM + n0 + l16];
    b.y = W[(bk + 1) * DIM + n0 + l16];
#pragma unroll
    for (int t = 0; t < NT; ++t) {
      const float* ap = As + (t * 16 + l16) * LDK + bk;
      v2f a;
      a.x = ap[0];
      a.y = ap[1];
      acc[t] = __builtin_amdgcn_wmma_f32_16x16x4_f32(
          false, a, false, b, (short)0, acc[t], false, false);
    }
  }
}

// Cooperative coalesced load of an A tile into padded LDS (zero-fill OOB rows).
template<int K, int NT>
__device__ __forceinline__ void load_tile(float* __restrict__ As,
                                          const float* __restrict__ A,
                                          int rowStart, int M) {
  constexpr int LDK = K + 4;
  constexpr int V4_PER_ROW = K / 4;
  constexpr int TOTAL = NT * 16 * V4_PER_ROW;
  for (int idx = threadIdx.x; idx < TOTAL; idx += blockDim.x) {
    int r  = idx / V4_PER_ROW;
    int c4 = idx % V4_PER_ROW;
    int grow = rowStart + r;
    float4 v = make_float4(0.f, 0.f, 0.f, 0.f);
    if (grow < M) v = *(const float4*)(A + (size_t)grow * K + c4 * 4);
    *(float4*)(&As[r * LDK + c4 * 4]) = v;
  }
}

__device__ __forceinline__ v8f zero8() {
  v8f z = {0.f, 0.f, 0.f, 0.f, 0.f, 0.f, 0.f, 0.f};
  return z;
}

// ---------------------------------------------------------------------------
// Fused 2-layer MLP: Out = relu(relu(A@W1+b1)@W2+b2), K2 == 128.
// Intermediate tile bounced through the same LDS buffer (stride 132).
// ---------------------------------------------------------------------------
template<int K1, int NT>
__global__ void __launch_bounds__(256)
fused_mlp_kernel(const float* __restrict__ A,
                 const float* __restrict__ W1, const float* __restrict__ b1,
                 const float* __restrict__ W2, const float* __restrict__ b2,
                 float* __restrict__ Out, int M) {
  __shared__ float As[NT * 16 * (K1 + 4)];
  const int rowStart = blockIdx.x * (NT * 16);
  const int lane = threadIdx.x & 31;
  const int wave = threadIdx.x >> 5;     // 8 waves -> 8 x 16 = 128 cols
  const int n0   = wave * 16;
  const int half = lane >> 4;
  const int l16  = lane & 15;
  const int col  = n0 + l16;

  load_tile<K1, NT>(As, A, rowStart, M);
  __syncthreads();

  v8f acc[NT];
#pragma unroll
  for (int t = 0; t < NT; ++t) acc[t] = zero8();
  mma_tiles<K1, NT>(As, W1, acc, n0, lane);

  __syncthreads();                       // everyone done reading layer-1 tile
  {
    const float bias1 = b1[col];
#pragma unroll
    for (int t = 0; t < NT; ++t)
#pragma unroll
      for (int i = 0; i < 8; ++i) {
        int lrow = t * 16 + half * 8 + i;
        As[lrow * (DIM + 4) + col] = fmaxf(acc[t][i] + bias1, 0.f);
      }
  }
  __syncthreads();

#pragma unroll
  for (int t = 0; t < NT; ++t) acc[t] = zero8();
  mma_tiles<DIM, NT>(As, W2, acc, n0, lane);

  const float bias2 = b2[col];
#pragma unroll
  for (int t = 0; t < NT; ++t)
#pragma unroll
    for (int i = 0; i < 8; ++i) {
      int grow = rowStart + t * 16 + half * 8 + i;
      if (grow < M)
        Out[(size_t)grow * DIM + col] = fmaxf(acc[t][i] + bias2, 0.f);
    }
}

// Plain GEMM: Out = A @ W, K = 128 (GCN transform; bias applied post-agg).
template<int NT>
__global__ void __launch_bounds__(256)
gemm_kernel(const float* __restrict__ A, const float* __restrict__ W,
            float* __restrict__ Out, int M) {
  __shared__ float As[NT * 16 * (DIM + 4)];
  const int rowStart = blockIdx.x * (NT * 16);
  const int lane = threadIdx.x & 31;
  const int wave = threadIdx.x >> 5;
  const int n0   = wave * 16;
  const int half = lane >> 4;
  const int l16  = lane & 15;
  const int col  = n0 + l16;

  load_tile<DIM, NT>(As, A, rowStart, M);
  __syncthreads();

  v8f acc[NT];
#pragma unroll
  for (int t = 0; t < NT; ++t) acc[t] = zero8();
  mma_tiles<DIM, NT>(As, W, acc, n0, lane);

#pragma unroll
  for (int t = 0; t < NT; ++t)
#pragma unroll
    for (int i = 0; i < 8; ++i) {
      int grow = rowStart + t * 16 + half * 8 + i;
      if (grow < M) Out[(size_t)grow * DIM + col] = acc[t][i];
    }
}

// ---------------------------------------------------------------------------
// GCN normalization / aggregation helpers
// ---------------------------------------------------------------------------
__global__ void zero_kernel(float* __restrict__ p, int n) {
  int i = blockIdx.x * blockDim.x + threadIdx.x;
  if (i < n) p[i] = 0.f;
}

__global__ void deg_kernel(const int* __restrict__ ei, float* __restrict__ deg, int E) {
  int e = blockIdx.x * blockDim.x + threadIdx.x;
  if (e < E) atomicAdd(&deg[ei[E + e]], 1.f);   // col = edge_index[1]
}

// in-place: dis[n] = rsqrt(deg[n] + 1)  (+1 = self loop; always > 0)
__global__ void rsqrt_kernel(float* __restrict__ d, int N) {
  int i = blockIdx.x * blockDim.x + threadIdx.x;
  if (i < N) d[i] = rsqrtf(d[i] + 1.f);
}

// agg[n,:] = dis[n]^2 * hw[n,:]   (self-loop term; fully overwrites agg)
__global__ void init_agg_kernel(const float* __restrict__ hw,
                                const float* __restrict__ dis,
                                float* __restrict__ agg, int N) {
  int idx = blockIdx.x * blockDim.x + threadIdx.x;   // over N*32 float4s
  if (idx >= N * 32) return;
  int n = idx >> 5;
  float d = dis[n];
  float s = d * d;
  float4 v = ((const float4*)hw)[idx];
  float4 o = make_float4(s * v.x, s * v.y, s * v.z, s * v.w);
  ((float4*)agg)[idx] = o;
}

// 32 lanes per edge: gather hw[row], scale, scatter-add into agg[col]
__global__ void scatter_kernel(const int* __restrict__ ei,
                               const float* __restrict__ hw,
                               const float* __restrict__ dis,
                               float* __restrict__ agg, int E) {
  int gtid = blockIdx.x * blockDim.x + threadIdx.x;
  int e = gtid >> 5;
  if (e >= E) return;
  int lane = gtid & 31;
  int r = ei[e];
  int c = ei[E + e];
  float nrm = dis[r] * dis[c];
  float4 v = *(const float4*)(hw + (size_t)r * DIM + lane * 4);
  float* dst = agg + (size_t)c * DIM + lane * 4;
  atomicAdd(dst + 0, nrm * v.x);
  atomicAdd(dst + 1, nrm * v.y);
  atomicAdd(dst + 2, nrm * v.z);
  atomicAdd(dst + 3, nrm * v.w);
}

template<bool RELU>
__global__ void bias_act_kernel(const float* __restrict__ agg,
                                const float* __restrict__ bias,
                                float* __restrict__ out, int N) {
  int idx = blockIdx.x * blockDim.x + threadIdx.x;
  if (idx >= N * DIM) return;
  float v = agg[idx] + bias[idx & (DIM - 1)];
  out[idx] = RELU ? fmaxf(v, 0.f) : v;
}

// ---------------------------------------------------------------------------
extern "C" void kernel_launch(void* const* d_in, const int* in_sizes, int n_in,
                              void* d_out, int out_size, void* d_ws, size_t ws_size,
                              hipStream_t stream) {
  const float* x   = (const float*)d_in[0];
  const int*   ei  = (const int*)d_in[1];
  const float* ea  = (const float*)d_in[2];
  const float* nW1 = (const float*)d_in[3];
  const float* nb1 = (const float*)d_in[4];
  const float* nW2 = (const float*)d_in[5];
  const float* nb2 = (const float*)d_in[6];
  const float* eW1 = (const float*)d_in[7];
  const float* eb1 = (const float*)d_in[8];
  const float* eW2 = (const float*)d_in[9];
  const float* eb2 = (const float*)d_in[10];
  const float* cW[3] = {(const float*)d_in[11], (const float*)d_in[13], (const float*)d_in[15]};
  const float* cb[3] = {(const float*)d_in[12], (const float*)d_in[14], (const float*)d_in[16]};

  const int N = in_sizes[0] / 256;   // 50000
  const int E = in_sizes[1] / 2;     // 800000

  float* out_h = (float*)d_out;                       // [N,128]
  float* out_e = out_h + (size_t)N * DIM;             // [E,128]

  float* hbuf   = (float*)d_ws;                       // [N,128]
  float* hwbuf  = hbuf   + (size_t)N * DIM;           // [N,128]
  float* aggbuf = hwbuf  + (size_t)N * DIM;           // [N,128]
  float* disbuf = aggbuf + (size_t)N * DIM;           // [N]

  // Node MLP (K1=256 -> 32-row tiles) and fused Edge MLP (K1=128 -> 64-row tiles)
  fused_mlp_kernel<256, 2><<<(N + 31) / 32, 256, 0, stream>>>(x, nW1, nb1, nW2, nb2, hbuf, N);
  fused_mlp_kernel<128, 4><<<(E + 63) / 64, 256, 0, stream>>>(ea, eW1, eb1, eW2, eb2, out_e, E);

  // Degree -> dis = rsqrt(deg + 1)
  zero_kernel<<<(N + 255) / 256, 256, 0, stream>>>(disbuf, N);
  deg_kernel<<<(E + 255) / 256, 256, 0, stream>>>(ei, disbuf, E);
  rsqrt_kernel<<<(N + 255) / 256, 256, 0, stream>>>(disbuf, N);

  // 3 GCN layers
  const float* hin = hbuf;
  for (int l = 0; l < 3; ++l) {
    gemm_kernel<4><<<(N + 63) / 64, 256, 0, stream>>>(hin, cW[l], hwbuf, N);
    init_agg_kernel<<<((N * 32) + 255) / 256, 256, 0, stream>>>(hwbuf, disbuf, aggbuf, N);
    scatter_kernel<<<(int)(((size_t)E * 32 + 255) / 256), 256, 0, stream>>>(ei, hwbuf, disbuf, aggbuf, E);
    if (l < 2) {
      bias_act_kernel<true><<<((N * DIM) + 255) / 256, 256, 0, stream>>>(aggbuf, cb[l], hbuf, N);
      hin = hbuf;
    } else {
      bias_act_kernel<false><<<((N * DIM) + 255) / 256, 256, 0, stream>>>(aggbuf, cb[l], out_h, N);
    }
  }
}